// PointNetSetAbstraction_72567767433503
// MI455X (gfx1250) — compile-verified
//
#include <hip/hip_runtime.h>

#define NPTS   4096
#define BATCH  8
#define S_OUT  1024
#define NSAMP  32
#define DIN    64
#define R2     0.04f
#define J_TOT  32768   // NSAMP * S_OUT columns per batch

typedef __attribute__((ext_vector_type(2))) float v2f;
typedef __attribute__((ext_vector_type(8))) float v8f;

// ---------------------------------------------------------------------------
// 1) Farthest point sampling: one block per batch, xyz resident in LDS.
//    Emits new_xyz (transposed, (B,3,S)) directly into d_out[0 .. 24575].
// ---------------------------------------------------------------------------
__global__ __launch_bounds__(256) void fps_kernel(const float* __restrict__ xyz,
                                                  float* __restrict__ out_newxyz) {
  __shared__ float xs[NPTS], ys[NPTS], zs[NPTS];
  __shared__ float rv[8];
  __shared__ int   ri[8];
  __shared__ int   sfar;
  const int b = blockIdx.x;
  const int t = threadIdx.x;
  const float* xb = xyz + (size_t)b * 3 * NPTS;
  for (int i = t; i < NPTS; i += 256) {
    xs[i] = xb[i];
    ys[i] = xb[NPTS + i];
    zs[i] = xb[2 * NPTS + i];
  }
  float dist[16];
#pragma unroll
  for (int i = 0; i < 16; ++i) dist[i] = 1e10f;
  if (t == 0) sfar = 0;
  __syncthreads();
  const int lane = t & 31, wave = t >> 5;
  for (int it = 0; it < S_OUT; ++it) {
    const int far = sfar;
    const float cx = xs[far], cy = ys[far], cz = zs[far];
    if (t == 0) {  // emit the centroid used this step (reference emits pre-update far)
      out_newxyz[(b * 3 + 0) * S_OUT + it] = cx;
      out_newxyz[(b * 3 + 1) * S_OUT + it] = cy;
      out_newxyz[(b * 3 + 2) * S_OUT + it] = cz;
    }
    float bestv = -1.0f; int besti = 0;
#pragma unroll
    for (int i = 0; i < 16; ++i) {
      const int n = t + i * 256;
      const float dx = xs[n] - cx, dy = ys[n] - cy, dz = zs[n] - cz;
      const float d  = dx * dx + dy * dy + dz * dz;
      const float nd = fminf(dist[i], d);
      dist[i] = nd;
      if (nd > bestv || (nd == bestv && n < besti)) { bestv = nd; besti = n; }
    }
    // wave32 argmax reduce (lowest index wins ties, matching jnp.argmax)
#pragma unroll
    for (int off = 16; off > 0; off >>= 1) {
      const float ov = __shfl_down(bestv, off);
      const int   oi = __shfl_down(besti, off);
      if (ov > bestv || (ov == bestv && oi < besti)) { bestv = ov; besti = oi; }
    }
    if (lane == 0) { rv[wave] = bestv; ri[wave] = besti; }
    __syncthreads();
    if (t == 0) {
      float bv = rv[0]; int bi = ri[0];
      for (int w = 1; w < 8; ++w)
        if (rv[w] > bv || (rv[w] == bv && ri[w] < bi)) { bv = rv[w]; bi = ri[w]; }
      sfar = bi;
    }
    __syncthreads();
  }
}

// ---------------------------------------------------------------------------
// 2) Ball query: one wave32 per center; ballot + prefix-popcount compaction
//    gives the first-32-ascending-index semantics exactly. gid layout [b][k][s].
// ---------------------------------------------------------------------------
__global__ __launch_bounds__(256) void query_kernel(const float* __restrict__ xyz,
                                                    const float* __restrict__ newxyz,
                                                    int* __restrict__ gid) {
  const int wave = threadIdx.x >> 5, lane = threadIdx.x & 31;
  const int gw = blockIdx.x * 8 + wave;
  const int b = gw >> 10;
  const int s = gw & 1023;
  const float* xb = xyz + (size_t)b * 3 * NPTS;
  const float cx = newxyz[(b * 3 + 0) * S_OUT + s];
  const float cy = newxyz[(b * 3 + 1) * S_OUT + s];
  const float cz = newxyz[(b * 3 + 2) * S_OUT + s];
  int* gb = gid + (size_t)b * NSAMP * S_OUT;
  int collected = 0;
  int first = -1;
  for (int c0 = 0; c0 < NPTS; c0 += 32) {
    const int n = c0 + lane;
    const float dx = xb[n] - cx, dy = xb[NPTS + n] - cy, dz = xb[2 * NPTS + n] - cz;
    const bool inr = (dx * dx + dy * dy + dz * dz) <= R2;
    const unsigned mask = __builtin_amdgcn_ballot_w32(inr);
    const int pos = collected + __popc(mask & ((1u << lane) - 1u));
    if (inr && pos < NSAMP) gb[pos * S_OUT + s] = n;
    if (first < 0 && mask) first = c0 + __builtin_ctz(mask);
    collected += __popc(mask);
    if (collected >= NSAMP) break;
  }
  const int filled = collected < NSAMP ? collected : NSAMP;
  if (lane >= filled && lane < NSAMP) gb[lane * S_OUT + s] = first;
}

// ---------------------------------------------------------------------------
// 3) Gather grouped features into bufA (layout [b][c(68)][j], c=67 is zero pad)
// ---------------------------------------------------------------------------
__global__ __launch_bounds__(256) void feat_kernel(const float* __restrict__ xyz,
                                                   const float* __restrict__ pts,
                                                   const float* __restrict__ newxyz,
                                                   const int* __restrict__ gid,
                                                   float* __restrict__ bufA) {
  const int bc     = blockIdx.x >> 7;                       // (b,c) pair
  const int within = ((blockIdx.x & 127) << 8) | threadIdx.x; // j = k*1024 + s
  const int b = bc / 68;
  const int c = bc % 68;
  const int k = within >> 10;
  const int s = within & 1023;
  const int idx = gid[((size_t)b * NSAMP + k) * S_OUT + s];
  float v;
  if (c < 3)       v = xyz[((size_t)b * 3 + c) * NPTS + idx] - newxyz[(b * 3 + c) * S_OUT + s];
  else if (c < 67) v = pts[((size_t)b * DIN + (c - 3)) * NPTS + idx];
  else             v = 0.0f;
  bufA[((size_t)b * 68 + c) * J_TOT + within] = v;
}

// Zero-pad w0 (64x67) into 64x68 so the WMMA K-loop has no divergence.
__global__ __launch_bounds__(256) void padw_kernel(const float* __restrict__ w,
                                                   float* __restrict__ wpad) {
  const int i = blockIdx.x * 256 + threadIdx.x;
  if (i >= 64 * 68) return;
  const int o = i / 68, c = i % 68;
  wpad[i] = (c < 67) ? w[o * 67 + c] : 0.0f;
}

// ---------------------------------------------------------------------------
// 4) GEMM via V_WMMA_F32_16X16X4_F32: out[o][j] = sum_c W[o][c]*in[c][j] + bias
//    One wave per 16-column strip; OT row-tiles of 16. Layouts per ISA 7.12.2.
// ---------------------------------------------------------------------------
template <int OT>
__global__ __launch_bounds__(256) void gemm_wmma_kernel(const float* __restrict__ in, int inC,
                                                        const float* __restrict__ w, int wK,
                                                        const float* __restrict__ bias,
                                                        float* __restrict__ out) {
  constexpr int O = OT * 16;
  const int lane = threadIdx.x & 31;
  const int gw = blockIdx.x * 8 + (threadIdx.x >> 5);
  const int b  = gw >> 11;            // 2048 strips per batch
  const int j0 = (gw & 2047) << 4;
  const int n  = lane & 15;
  const int hi = lane >> 4;
  const float* inb = in + (size_t)b * inC * J_TOT;

  v8f zero = {0.f, 0.f, 0.f, 0.f, 0.f, 0.f, 0.f, 0.f};
  v8f acc[OT];
#pragma unroll
  for (int t = 0; t < OT; ++t) acc[t] = zero;

  for (int kc = 0; kc < wK; kc += 4) {
    const int r0 = kc + 2 * hi;       // lanes 0-15: K0/K1, lanes 16-31: K2/K3
    v2f bf;
    bf.x = inb[(size_t)r0 * J_TOT + j0 + n];
    bf.y = inb[(size_t)(r0 + 1) * J_TOT + j0 + n];
#pragma unroll
    for (int t = 0; t < OT; ++t) {
      const int o = t * 16 + n;       // A row m = lane & 15
      v2f af;
      af.x = w[o * wK + r0];
      af.y = w[o * wK + r0 + 1];
      acc[t] = __builtin_amdgcn_wmma_f32_16x16x4_f32(
          false, af, false, bf, (short)0, acc[t], false, false);
    }
  }

  float* outb = out + (size_t)b * O * J_TOT;
#pragma unroll
  for (int t = 0; t < OT; ++t) {
#pragma unroll
    for (int v = 0; v < 8; ++v) {
      const int m = v + 8 * hi;       // C/D layout: m = vgpr + 8*(lane>>4)
      const int o = t * 16 + m;
      outb[(size_t)o * J_TOT + j0 + n] = acc[t][v] + bias[o];
    }
  }
}

// ---------------------------------------------------------------------------
// 5) BatchNorm stats: one block per channel over (B, 32768) -> alpha/beta
// ---------------------------------------------------------------------------
__global__ __launch_bounds__(256) void stats_kernel(const float* __restrict__ y, int O,
                                                    const float* __restrict__ g,
                                                    const float* __restrict__ be,
                                                    float* __restrict__ alpha,
                                                    float* __restrict__ beta) {
  const int o = blockIdx.x;
  const int t = threadIdx.x;
  double s = 0.0, sq = 0.0;
  for (int b = 0; b < BATCH; ++b) {
    const float* p = y + ((size_t)b * O + o) * J_TOT;
    for (int j = t; j < J_TOT; j += 256) {
      const float v = p[j];
      s += v; sq += (double)v * v;
    }
  }
  __shared__ double ss[256], ssq[256];
  ss[t] = s; ssq[t] = sq;
  __syncthreads();
  for (int off = 128; off > 0; off >>= 1) {
    if (t < off) { ss[t] += ss[t + off]; ssq[t] += ssq[t + off]; }
    __syncthreads();
  }
  if (t == 0) {
    const double N = (double)BATCH * J_TOT;
    const double m = ss[0] / N;
    const double var = ssq[0] / N - m * m;
    const float a = g[o] / sqrtf((float)var + 1e-5f);
    alpha[o] = a;
    beta[o]  = be[o] - (float)m * a;
  }
}

__global__ __launch_bounds__(256) void normrelu_kernel(float* __restrict__ y, int O,
                                                       const float* __restrict__ alpha,
                                                       const float* __restrict__ beta) {
  const size_t i = (size_t)blockIdx.x * 256 + threadIdx.x;
  const int o = (int)(i >> 15) % O;   // J_TOT == 1<<15
  float v = y[i];
  v = alpha[o] * v + beta[o];
  y[i] = v > 0.0f ? v : 0.0f;
}

// 6) Max over K samples -> d_out[24576 ...]
__global__ __launch_bounds__(256) void maxpool_kernel(const float* __restrict__ y,
                                                      float* __restrict__ out) {
  const int i = blockIdx.x * 256 + threadIdx.x;  // (b*128+o)*1024 + s
  const int s = i & 1023;
  const int bo = i >> 10;
  const float* p = y + (size_t)bo * J_TOT;
  float m = -1e30f;
#pragma unroll
  for (int k = 0; k < NSAMP; ++k) m = fmaxf(m, p[k * S_OUT + s]);
  out[24576 + i] = m;
}

extern "C" void kernel_launch(void* const* d_in, const int* in_sizes, int n_in,
                              void* d_out, int out_size, void* d_ws, size_t ws_size,
                              hipStream_t stream) {
  const float* xyz = (const float*)d_in[0];
  const float* pts = (const float*)d_in[1];
  const float* w0  = (const float*)d_in[2];
  const float* b0  = (const float*)d_in[3];
  const float* g0  = (const float*)d_in[4];
  const float* be0 = (const float*)d_in[5];
  const float* w1  = (const float*)d_in[6];
  const float* b1  = (const float*)d_in[7];
  const float* g1  = (const float*)d_in[8];
  const float* be1 = (const float*)d_in[9];
  const float* w2  = (const float*)d_in[10];
  const float* b2  = (const float*)d_in[11];
  const float* g2  = (const float*)d_in[12];
  const float* be2 = (const float*)d_in[13];
  float* out = (float*)d_out;
  char* ws = (char*)d_ws;

  // workspace layout
  int*   gid   = (int*)ws;                                  // 8*32*1024*4   = 1 MB
  float* alpha = (float*)(ws + 1048576);                    // 128 f
  float* beta  = alpha + 128;                               // 128 f
  float* wpad0 = beta + 128;                                // 64*68 f
  float* bufA  = (float*)(ws + 1067008);                    // 8*68*32768 f  = 71.3 MB
  float* bufB  = (float*)(ws + 1067008 + 71303168ull);      // 8*128*32768 f = 134.2 MB
  (void)in_sizes; (void)n_in; (void)out_size; (void)ws_size;

  fps_kernel<<<8, 256, 0, stream>>>(xyz, out);
  query_kernel<<<1024, 256, 0, stream>>>(xyz, out, gid);
  feat_kernel<<<69632, 256, 0, stream>>>(xyz, pts, out, gid, bufA);
  padw_kernel<<<17, 256, 0, stream>>>(w0, wpad0);

  // layer 0: 68(padded K) -> 64
  gemm_wmma_kernel<4><<<2048, 256, 0, stream>>>(bufA, 68, wpad0, 68, b0, bufB);
  stats_kernel<<<64, 256, 0, stream>>>(bufB, 64, g0, be0, alpha, beta);
  normrelu_kernel<<<65536, 256, 0, stream>>>(bufB, 64, alpha, beta);

  // layer 1: 64 -> 64
  gemm_wmma_kernel<4><<<2048, 256, 0, stream>>>(bufB, 64, w1, 64, b1, bufA);
  stats_kernel<<<64, 256, 0, stream>>>(bufA, 64, g1, be1, alpha, beta);
  normrelu_kernel<<<65536, 256, 0, stream>>>(bufA, 64, alpha, beta);

  // layer 2: 64 -> 128
  gemm_wmma_kernel<8><<<2048, 256, 0, stream>>>(bufA, 64, w2, 64, b2, bufB);
  stats_kernel<<<128, 256, 0, stream>>>(bufB, 128, g2, be2, alpha, beta);
  normrelu_kernel<<<131072, 256, 0, stream>>>(bufB, 128, alpha, beta);

  maxpool_kernel<<<4096, 256, 0, stream>>>(bufB, out);
}